// WeightCroLoss_47373489275465
// MI455X (gfx1250) — compile-verified
//
#include <hip/hip_runtime.h>
#include <hip/hip_bf16.h>

typedef __attribute__((ext_vector_type(4))) float v4f;
typedef __attribute__((ext_vector_type(4))) int   v4i;
typedef __attribute__((ext_vector_type(2))) float v2f;
typedef __attribute__((ext_vector_type(8))) float v8f;

// -----------------------------------------------------------------------------
// Wave32 all-lane sum using V_WMMA_F32_16X16X4_F32 (CDNA5).
// B = all-ones makes B's lane layout irrelevant: D[m][n] = sum_k A[m][k].
// A layout (32-bit 16x4): vgpr0: lanes0-15 -> A[m][0], lanes16-31 -> A[m][2]
//                         vgpr1: lanes0-15 -> A[m][1], lanes16-31 -> A[m][3]
// D layout (32-bit 16x16): vgpr r: lanes0-15 -> D[r][n], lanes16-31 -> D[8+r][n]
//
// Pass 1: A = (v, 0)  ->  D1[m][*] = v_m + v_{m+16}  =: s_m  (const per row)
// Pass 2 (x4, C-accumulating): A = (d1[2i], d1[2i+1]) sums rows
//   {2i, 2i+1, 8+2i, 9+2i}; four calls cover all 16 rows -> total sum,
//   broadcast into every element of the accumulator.
// Requires EXEC == all 1s (call from a fully-active wave only).
// -----------------------------------------------------------------------------
__device__ __forceinline__ float wmma_wave_allsum(float v) {
    v2f ones; ones[0] = 1.0f; ones[1] = 1.0f;
    v2f a;    a[0] = v;       a[1] = 0.0f;
    v8f zero = {};
    v8f d1 = __builtin_amdgcn_wmma_f32_16x16x4_f32(
        false, a, false, ones, (short)0, zero, false, false);

    v8f acc = {};
    v2f a2;
    a2[0] = d1[0]; a2[1] = d1[1];
    acc = __builtin_amdgcn_wmma_f32_16x16x4_f32(
        false, a2, false, ones, (short)0, acc, false, false);
    a2[0] = d1[2]; a2[1] = d1[3];
    acc = __builtin_amdgcn_wmma_f32_16x16x4_f32(
        false, a2, false, ones, (short)0, acc, false, false);
    a2[0] = d1[4]; a2[1] = d1[5];
    acc = __builtin_amdgcn_wmma_f32_16x16x4_f32(
        false, a2, false, ones, (short)0, acc, false, false);
    a2[0] = d1[6]; a2[1] = d1[7];
    acc = __builtin_amdgcn_wmma_f32_16x16x4_f32(
        false, a2, false, ones, (short)0, acc, false, false);
    return acc[0];
}

// Stable BCE-with-logits element: max(z,0) - t*z + log(1 + exp(-|z|))
__device__ __forceinline__ float bce_elem(float z, float t) {
    float az = __builtin_fabsf(z);
    float e  = __expf(-az);          // v_exp_f32 (underflows to 0 for large |z|)
    float l  = __logf(1.0f + e);     // v_log_f32
    return fmaxf(z, 0.0f) - t * z + l;
}

#define NTHR 256
#define NBLK 4096

__global__ __launch_bounds__(NTHR)
void bce_partial_kernel(const int* __restrict__ pred,
                        const float* __restrict__ labels,
                        float* __restrict__ partial,
                        long long n4) {
    const long long tid    = (long long)blockIdx.x * NTHR + threadIdx.x;
    const long long stride = (long long)gridDim.x * NTHR;

    const v4f* __restrict__ zp = (const v4f*)labels;
    const v4i* __restrict__ tp = (const v4i*)pred;

    float acc = 0.0f;
    for (long long i = tid; i < n4; i += stride) {
        // Non-temporal b128 streaming loads: 536 MB touched once, don't
        // thrash the 192 MB L2.
        v4f z = __builtin_nontemporal_load(zp + i);
        v4i t = __builtin_nontemporal_load(tp + i);
#pragma unroll
        for (int c = 0; c < 4; ++c) {
            acc += bce_elem(z[c], (float)t[c]);
        }
    }

    __shared__ float smem[NTHR];
    smem[threadIdx.x] = acc;
    __syncthreads();

    // Wave 0 (lanes 0..31, fully active -> EXEC all-1s) folds 8 waves' worth
    // of partials, then does the WMMA cross-lane sum.
    if (threadIdx.x < 32) {
        float s = 0.0f;
#pragma unroll
        for (int j = 0; j < NTHR / 32; ++j) s += smem[threadIdx.x + 32 * j];
        float tot = wmma_wave_allsum(s);
        if (threadIdx.x == 0) partial[blockIdx.x] = tot;
    }
}

__global__ __launch_bounds__(NTHR)
void bce_final_kernel(const float* __restrict__ partial,
                      float* __restrict__ out,
                      int n) {
    float acc = 0.0f;
    for (int i = threadIdx.x; i < n; i += NTHR) acc += partial[i];

    __shared__ float smem[NTHR];
    smem[threadIdx.x] = acc;
    __syncthreads();

    if (threadIdx.x < 32) {
        float s = 0.0f;
#pragma unroll
        for (int j = 0; j < NTHR / 32; ++j) s += smem[threadIdx.x + 32 * j];
        float tot = wmma_wave_allsum(s);
        if (threadIdx.x == 0) out[0] = tot;
    }
}

extern "C" void kernel_launch(void* const* d_in, const int* in_sizes, int n_in,
                              void* d_out, int out_size, void* d_ws, size_t ws_size,
                              hipStream_t stream) {
    const int*   pred   = (const int*)d_in[0];    // int32 0/1 targets
    const float* labels = (const float*)d_in[1];  // fp32 logits

    const long long n  = (long long)in_sizes[0];  // 16384*4096 = 67,108,864
    const long long n4 = n >> 2;                  // vec4 groups

    float* partials = (float*)d_ws;               // NBLK * 4 = 16 KB scratch

    bce_partial_kernel<<<NBLK, NTHR, 0, stream>>>(pred, labels, partials, n4);
    bce_final_kernel<<<1, NTHR, 0, stream>>>(partials, (float*)d_out, NBLK);
}